// HypergraphConv_35751307772368
// MI455X (gfx1250) — compile-verified
//
#include <hip/hip_runtime.h>
#include <hip/hip_bf16.h>

#define IN_DIM  128
#define OUT_DIM 128

typedef __attribute__((ext_vector_type(2))) float v2f;
typedef __attribute__((ext_vector_type(8))) float v8f;

// ---------------------------------------------------------------------------
// Phase 1: m[c,:] += dv[r]^-1/2 * X[r,:]  for each incidence entry (r,c).
// One wave (32 lanes) per NNZ entry; each lane owns a 4-float slice of the
// 128-wide row -> coalesced float4 gather + 4 global_atomic_add_f32.
// The entry index is wave-uniform: readfirstlane pins it to an SGPR so the
// index / degree loads become scalar (s_load) broadcasts, not 32 VMEM loads.
// ---------------------------------------------------------------------------
__global__ __launch_bounds__(256) void hgc_scatter_edges(
    const float* __restrict__ X, const int* __restrict__ rows,
    const int* __restrict__ cols, const float* __restrict__ dv,
    float* __restrict__ m, int nnz) {
  long tid = (long)blockIdx.x * blockDim.x + threadIdx.x;
  int e = __builtin_amdgcn_readfirstlane((int)(tid >> 5));  // wave-uniform -> SGPR
  if (e >= nnz) return;                                     // scalar branch
  int lane = (int)(tid & 31);
  int r = rows[e];                                          // s_load
  int c = cols[e];                                          // s_load
  float s = __frsqrt_rn(dv[r]);                             // dv^{-1/2}, scalar fetch
  const float4 xv = *(const float4*)(X + (long)r * IN_DIM + lane * 4);
  float* dst = m + (long)c * IN_DIM + lane * 4;
  unsafeAtomicAdd(dst + 0, s * xv.x);
  unsafeAtomicAdd(dst + 1, s * xv.y);
  unsafeAtomicAdd(dst + 2, s * xv.z);
  unsafeAtomicAdd(dst + 3, s * xv.w);
}

// ---------------------------------------------------------------------------
// Phase 2: y[r,:] += dv[r]^-1/2 * de[c]^-1 * m[c,:].  Both diagonal scalings
// folded into the gather-scatter (dv scale depends only on destination row r).
// ---------------------------------------------------------------------------
__global__ __launch_bounds__(256) void hgc_scatter_nodes(
    const float* __restrict__ m, const int* __restrict__ rows,
    const int* __restrict__ cols, const float* __restrict__ dv,
    const float* __restrict__ de, float* __restrict__ y, int nnz) {
  long tid = (long)blockIdx.x * blockDim.x + threadIdx.x;
  int e = __builtin_amdgcn_readfirstlane((int)(tid >> 5));  // wave-uniform -> SGPR
  if (e >= nnz) return;
  int lane = (int)(tid & 31);
  int r = rows[e];
  int c = cols[e];
  float s = __frsqrt_rn(dv[r]) * __frcp_rn(de[c]);          // dv^{-1/2} * de^{-1}
  const float4 mv = *(const float4*)(m + (long)c * IN_DIM + lane * 4);
  float* dst = y + (long)r * IN_DIM + lane * 4;
  unsafeAtomicAdd(dst + 0, s * mv.x);
  unsafeAtomicAdd(dst + 1, s * mv.y);
  unsafeAtomicAdd(dst + 2, s * mv.z);
  unsafeAtomicAdd(dst + 3, s * mv.w);
}

// ---------------------------------------------------------------------------
// Phase 3: out = y @ W, fp32, in place on y (each block only touches its own
// 16 rows). 8 waves per block: wave w computes the 16x16 tile at columns
// [16w, 16w+16) via 32 chained V_WMMA_F32_16X16X4_F32 (K = 128).
//
// Operand packing (ISA 7.12.2, 32-bit layouts):
//   A (16x4):  lane L -> row M = L%16;  VGPR v -> K = k0 + 2*(L/16) + v
//   B (4x16):  lane L -> col N = L%16;  VGPR v -> K = k0 + 2*(L/16) + v
//   C/D:       lane L, VGPR j -> (M = j + 8*(L/16), N = L%16)
//
// LDS: W staged in two 64-row halves (32 KB), Y tile padded to stride 132
// so the 16 A-lanes reading a column of the tile hit 16 distinct banks.
// ---------------------------------------------------------------------------
#define Y_STRIDE 132  // 128 + 4 pad -> bank = (l16*132 + k) % 64 distinct per l16

__global__ __launch_bounds__(256) void hgc_gemm_wmma(
    float* __restrict__ y, const float* __restrict__ W, int N) {
  __shared__ float Wl[64 * OUT_DIM];        // 32 KB: one 64-row half of W
  __shared__ float Yl[16 * Y_STRIDE];       // ~8.4 KB padded 16x128 tile

  const int t = threadIdx.x;
  const int rowBase = blockIdx.x * 16;

  // Stage the block's 16 rows of y into LDS (zero-fill past N).
  for (int i = 0; i < 2; ++i) {
    int idx = (i * 256 + t) * 4;            // [0, 2048)
    int rr = idx >> 7;                      // row in tile
    int cc = idx & 127;                     // col
    float4 v = make_float4(0.f, 0.f, 0.f, 0.f);
    int row = rowBase + rr;
    if (row < N) v = *(const float4*)(y + (long)row * IN_DIM + cc);
    *(float4*)(Yl + rr * Y_STRIDE + cc) = v;
  }

  const int lane = t & 31;
  const int wave = t >> 5;                  // 0..7 -> column tile
  const int half = lane >> 4;               // hi/lo 16 lanes
  const int l16  = lane & 15;
  const int cb   = wave * 16;               // column base of this wave's tile

  v8f acc = {};                             // fp32 16x16 accumulator (8 VGPRs)

  for (int kb = 0; kb < IN_DIM; kb += 64) {
    __syncthreads();                        // prior-phase LDS reads done
    // Stage W[kb .. kb+63][:] : 8192 floats / 256 threads = 8 float4 each.
    for (int i = 0; i < 8; ++i) {
      int idx = (i * 256 + t) * 4;          // [0, 8192)
      *(float4*)(Wl + idx) = *(const float4*)(W + (long)kb * OUT_DIM + idx);
    }
    __syncthreads();

    #pragma unroll
    for (int k0 = 0; k0 < 64; k0 += 4) {
      const int ka = k0 + 2 * half;         // this half-wave's K pair
      v2f a, b;
      a.x = Yl[l16 * Y_STRIDE + kb + ka];
      a.y = Yl[l16 * Y_STRIDE + kb + ka + 1];
      b.x = Wl[ka * OUT_DIM + cb + l16];
      b.y = Wl[(ka + 1) * OUT_DIM + cb + l16];
      acc = __builtin_amdgcn_wmma_f32_16x16x4_f32(
          /*neg_a=*/false, a, /*neg_b=*/false, b,
          /*c_mod=*/(short)0, acc, /*reuse_a=*/false, /*reuse_b=*/false);
    }
  }

  // Write D: VGPR j holds row (j + 8*half) of the tile, column cb + l16.
  #pragma unroll
  for (int j = 0; j < 8; ++j) {
    int row = rowBase + j + 8 * half;
    if (row < N) y[(long)row * OUT_DIM + cb + l16] = acc[j];
  }
}

// ---------------------------------------------------------------------------
extern "C" void kernel_launch(void* const* d_in, const int* in_sizes, int n_in,
                              void* d_out, int out_size, void* d_ws, size_t ws_size,
                              hipStream_t stream) {
  const float* X    = (const float*)d_in[0];
  const int*   rows = (const int*)d_in[1];
  const int*   cols = (const int*)d_in[2];
  const float* dv   = (const float*)d_in[3];
  const float* de   = (const float*)d_in[4];
  const float* W    = (const float*)d_in[5];
  float*       out  = (float*)d_out;

  const int nnz = in_sizes[1];
  const int N   = in_sizes[3];
  const int E   = in_sizes[4];

  float* m = (float*)d_ws;                  // E x 128 edge accumulator

  // Zero the two accumulators (graph-capture-safe stream memsets).
  hipMemsetAsync(m, 0, (size_t)E * IN_DIM * sizeof(float), stream);
  hipMemsetAsync(out, 0, (size_t)N * IN_DIM * sizeof(float), stream);

  // One wave per incidence entry.
  long scatterThreads = (long)nnz * 32;
  int scatterBlocks = (int)((scatterThreads + 255) / 256);
  hgc_scatter_edges<<<scatterBlocks, 256, 0, stream>>>(X, rows, cols, dv, m, nnz);
  hgc_scatter_nodes<<<scatterBlocks, 256, 0, stream>>>(m, rows, cols, dv, de, out, nnz);

  // In-place WMMA GEMM: out = out @ W.
  int gemmBlocks = (N + 15) / 16;
  hgc_gemm_wmma<<<gemmBlocks, 256, 0, stream>>>(out, W, N);
}